// KGPrompt_19507741458967
// MI455X (gfx1250) — compile-verified
//
#include <hip/hip_runtime.h>
#include <math.h>

// ---------------- problem constants ----------------
#define Hh   512
#define TH   768
#define NH   8
#define NL   12
#define NB   2
#define NE   16000
#define NU   2000
#define NI   4000
#define NNZ  40000
#define KK   10
#define Bb   64
#define MD   256
#define E_TOT (NNZ + NU*KK)          // 60000
#define N_MAP (Hh*Hh)                // 262144
#define P2N  (NL*NB*Hh)              // 12288
#define PROMPT_N (NL*NB*Bb*NH*(Hh/NH))   // 786432
#define GAMA 0.1f

typedef __attribute__((ext_vector_type(2))) float v2f;
typedef __attribute__((ext_vector_type(8))) float v8f;

// ================= generic WMMA f32 GEMM =================
// C[M,N] = epilogue(A[M,K] @ B[K,N] (+bias))
// block = 128 threads (4 wave32s); each wave computes a 16x64 strip.
// BTRANS/GATHER are compile-time so the K-loop is branch-free with
// pure pointer-marching address math (no 64-bit muls in the loop).
// modes (runtime, epilogue only): 0 plain, 1 relu, 2 sigmoid,
//                                 3 prelu(x)*extra, 4 x+extra (residual)
template <int BTRANS, int GATHER>
__global__ __launch_bounds__(128)
void wmma_gemm_kernel(const float* __restrict__ A, const float* __restrict__ Bm,
                      float* __restrict__ C,
                      int M, int N, int Kd, int lda, int ldb, int ldc,
                      const float* __restrict__ bias,
                      const int* __restrict__ a_gather,
                      int mode, const float* __restrict__ extra, int lde,
                      const float* __restrict__ prelu_a_ptr)
{
    const int lane = threadIdx.x & 31;
    const int wave = threadIdx.x >> 5;
    const int half = lane >> 4;
    const int l    = lane & 15;
    const int n0   = blockIdx.x * 64;
    const int m0   = blockIdx.y * 64 + wave * 16;

    int arow = m0 + l; if (arow >= M) arow = M - 1;           // clamp (stores guarded)
    int arow_g = arow;
    if (GATHER) arow_g = a_gather[arow];
    const float* aptr = A + (size_t)arow_g * lda + 2 * half;

    // per-subtile B pointers, clamped once (columns are K-invariant)
    int nc0 = n0 +  0 + l; if (nc0 >= N) nc0 = N - 1;
    int nc1 = n0 + 16 + l; if (nc1 >= N) nc1 = N - 1;
    int nc2 = n0 + 32 + l; if (nc2 >= N) nc2 = N - 1;
    int nc3 = n0 + 48 + l; if (nc3 >= N) nc3 = N - 1;
    const size_t boff = BTRANS ? (size_t)(2 * half) : (size_t)(2 * half) * ldb;
    const float* p0 = BTRANS ? (Bm + (size_t)nc0 * ldb + boff) : (Bm + boff + nc0);
    const float* p1 = BTRANS ? (Bm + (size_t)nc1 * ldb + boff) : (Bm + boff + nc1);
    const float* p2 = BTRANS ? (Bm + (size_t)nc2 * ldb + boff) : (Bm + boff + nc2);
    const float* p3 = BTRANS ? (Bm + (size_t)nc3 * ldb + boff) : (Bm + boff + nc3);
    const size_t bstep = BTRANS ? (size_t)4 : (size_t)4 * ldb;
    const size_t bhi   = BTRANS ? (size_t)1 : (size_t)ldb;     // +1 K within fragment

    v8f acc0, acc1, acc2, acc3;
#pragma unroll
    for (int j = 0; j < 8; ++j) { acc0[j] = 0.f; acc1[j] = 0.f; acc2[j] = 0.f; acc3[j] = 0.f; }

    for (int k0 = 0; k0 < Kd; k0 += 4) {
        v2f a; a.x = aptr[0]; a.y = aptr[1]; aptr += 4;
        v2f b0; b0.x = p0[0]; b0.y = p0[bhi]; p0 += bstep;
        v2f b1; b1.x = p1[0]; b1.y = p1[bhi]; p1 += bstep;
        v2f b2; b2.x = p2[0]; b2.y = p2[bhi]; p2 += bstep;
        v2f b3; b3.x = p3[0]; b3.y = p3[bhi]; p3 += bstep;
        acc0 = __builtin_amdgcn_wmma_f32_16x16x4_f32(false, a, false, b0, (short)0, acc0, false, false);
        acc1 = __builtin_amdgcn_wmma_f32_16x16x4_f32(false, a, false, b1, (short)0, acc1, false, false);
        acc2 = __builtin_amdgcn_wmma_f32_16x16x4_f32(false, a, false, b2, (short)0, acc2, false, false);
        acc3 = __builtin_amdgcn_wmma_f32_16x16x4_f32(false, a, false, b3, (short)0, acc3, false, false);
    }

    const float pa = prelu_a_ptr ? *prelu_a_ptr : 0.0f;
    v8f accs[4] = {acc0, acc1, acc2, acc3};
#pragma unroll
    for (int t = 0; t < 4; ++t) {
        const int n = n0 + 16 * t + l;
#pragma unroll
        for (int v = 0; v < 8; ++v) {
            const int row = m0 + v + (half ? 8 : 0);
            if (row < M && n < N) {
                float x = accs[t][v];
                if (bias) x += bias[n];
                if (mode == 1)      x = fmaxf(x, 0.0f);
                else if (mode == 2) x = 1.0f / (1.0f + expf(-x));
                else if (mode == 3) { float g = extra[(size_t)row * lde + n];
                                      x = (x >= 0.0f ? x : pa * x) * g; }
                else if (mode == 4) x += extra[(size_t)row * lde + n];
                C[(size_t)row * ldc + n] = x;
            }
        }
    }
}

// ========== fused hypernet: enhanced[b,k] = sum_h ue[b,h]*(hf@W2 + b2)[b,h*512+k] ==========
// grid 4096 blocks; block owns N-cols [64*blk, +64) of the 262144-wide mapping,
// i.e. a unique (h, k0) pair -> W2 streamed exactly once, mapping never hits memory.
__global__ __launch_bounds__(128)
void hypernet_kernel(const float* __restrict__ hf,    // [64,256]
                     const float* __restrict__ W2,    // [256,262144]
                     const float* __restrict__ b2,    // [262144]
                     const float* __restrict__ ue,    // [64,512] gathered user_embedding
                     float* __restrict__ enhanced)    // [64,512] (pre-zeroed)
{
    const int lane = threadIdx.x & 31;
    const int wave = threadIdx.x >> 5;
    const int half = lane >> 4;
    const int l    = lane & 15;
    const int n0   = blockIdx.x * 64;
    const int h    = n0 >> 9;            // which hidden row of the mapping
    const int m0   = wave * 16;          // batch rows (B=64 total)

    const float* aptr = hf + (size_t)(m0 + l) * MD + 2 * half;
    const size_t base = (size_t)(2 * half) * N_MAP;
    const float* p0 = W2 + base + (n0 +  0 + l);
    const float* p1 = W2 + base + (n0 + 16 + l);
    const float* p2 = W2 + base + (n0 + 32 + l);
    const float* p3 = W2 + base + (n0 + 48 + l);
    const size_t bstep = (size_t)4 * N_MAP;

    v8f acc0, acc1, acc2, acc3;
#pragma unroll
    for (int j = 0; j < 8; ++j) { acc0[j] = 0.f; acc1[j] = 0.f; acc2[j] = 0.f; acc3[j] = 0.f; }

    for (int k0 = 0; k0 < MD; k0 += 4) {
        v2f a; a.x = aptr[0]; a.y = aptr[1]; aptr += 4;
        __builtin_prefetch(p0 + 2 * bstep, 0, 1);              // stream-ahead hint
        v2f b0; b0.x = p0[0]; b0.y = p0[N_MAP]; p0 += bstep;
        v2f b1; b1.x = p1[0]; b1.y = p1[N_MAP]; p1 += bstep;
        v2f b2; b2.x = p2[0]; b2.y = p2[N_MAP]; p2 += bstep;
        v2f b3; b3.x = p3[0]; b3.y = p3[N_MAP]; p3 += bstep;
        acc0 = __builtin_amdgcn_wmma_f32_16x16x4_f32(false, a, false, b0, (short)0, acc0, false, false);
        acc1 = __builtin_amdgcn_wmma_f32_16x16x4_f32(false, a, false, b1, (short)0, acc1, false, false);
        acc2 = __builtin_amdgcn_wmma_f32_16x16x4_f32(false, a, false, b2, (short)0, acc2, false, false);
        acc3 = __builtin_amdgcn_wmma_f32_16x16x4_f32(false, a, false, b3, (short)0, acc3, false, false);
    }

    v8f accs[4] = {acc0, acc1, acc2, acc3};
#pragma unroll
    for (int t = 0; t < 4; ++t) {
        const int n = n0 + 16 * t + l;
        const int k = n & (Hh - 1);
#pragma unroll
        for (int v = 0; v < 8; ++v) {
            const int row = m0 + v + (half ? 8 : 0);
            float x = accs[t][v] + b2[n];
            x *= ue[(size_t)row * Hh + h];
            atomicAdd(&enhanced[(size_t)row * Hh + k], x);
        }
    }
}

// ================= small helper kernels =================
__global__ void group_mean_kernel(const int* __restrict__ sys,
                                  const float* __restrict__ tok, float* __restrict__ gm) {
    const int b = blockIdx.x; const int sid = sys[b];
    int cnt = 0;
    for (int j = 0; j < Bb; ++j) cnt += (sys[j] == sid);
    const float inv = 1.0f / (float)cnt;
    for (int hcol = threadIdx.x; hcol < Hh; hcol += blockDim.x) {
        float s = 0.0f;
        for (int j = 0; j < Bb; ++j) if (sys[j] == sid) s += tok[j * Hh + hcol];
        gm[b * Hh + hcol] = s * inv;
    }
}

__global__ void gather_rows_kernel(const float* __restrict__ src, const int* __restrict__ idx,
                                   float* __restrict__ dst, long long n, int w) {
    long long i = (long long)blockIdx.x * blockDim.x + threadIdx.x;
    if (i >= n) return;
    int r = (int)(i / w);
    dst[i] = src[(size_t)idx[r] * w + (int)(i % w)];
}

__global__ void scatter_rows_kernel(const int* __restrict__ sys,
                                    const float* __restrict__ enh, const float* __restrict__ gm,
                                    float* __restrict__ all_tok, float* __restrict__ tok_edge) {
    int i = blockIdx.x * blockDim.x + threadIdx.x;
    if (i >= Bb * Hh) return;
    int b = i >> 9, hcol = i & (Hh - 1);
    int row = sys[b];
    all_tok[(size_t)row * Hh + hcol]  = enh[i];
    tok_edge[(size_t)row * Hh + hcol] = gm[i];
}

__global__ void rownorm_kernel(float* __restrict__ X) {
    const int row = blockIdx.x;
    __shared__ float red[256];
    float s = 0.0f;
    for (int h = threadIdx.x; h < Hh; h += 256) { float x = X[(size_t)row * Hh + h]; s += x * x; }
    red[threadIdx.x] = s; __syncthreads();
    for (int st = 128; st > 0; st >>= 1) {
        if (threadIdx.x < st) red[threadIdx.x] += red[threadIdx.x + st];
        __syncthreads();
    }
    const float sc = 1.0f / fmaxf(sqrtf(red[0]), 1e-12f);
    for (int h = threadIdx.x; h < Hh; h += 256) X[(size_t)row * Hh + h] *= sc;
}

__global__ void topk_kernel(const float* __restrict__ sim,
                            float* __restrict__ tv, int* __restrict__ ti) {
    const int u = blockIdx.x;
    __shared__ float sv[256]; __shared__ int si[256]; __shared__ int chosen[KK];
    const float* row = sim + (size_t)u * NI;
    for (int t = 0; t < KK; ++t) {
        float best = -1e30f; int bi = 0x7fffffff;
        for (int i = threadIdx.x; i < NI; i += 256) {
            bool used = false;
            for (int q = 0; q < t; ++q) if (chosen[q] == i) used = true;
            if (used) continue;
            float vv = row[i];
            if (vv > best || (vv == best && i < bi)) { best = vv; bi = i; }
        }
        sv[threadIdx.x] = best; si[threadIdx.x] = bi; __syncthreads();
        for (int st = 128; st > 0; st >>= 1) {
            if (threadIdx.x < st) {
                if (sv[threadIdx.x + st] > sv[threadIdx.x] ||
                    (sv[threadIdx.x + st] == sv[threadIdx.x] && si[threadIdx.x + st] < si[threadIdx.x])) {
                    sv[threadIdx.x] = sv[threadIdx.x + st]; si[threadIdx.x] = si[threadIdx.x + st];
                }
            }
            __syncthreads();
        }
        if (threadIdx.x == 0) { chosen[t] = si[0]; tv[u * KK + t] = sv[0]; ti[u * KK + t] = si[0]; }
        __syncthreads();
    }
}

__global__ void edges_kernel(const int* __restrict__ ar, const int* __restrict__ ac,
                             const float* __restrict__ av, const int* __restrict__ itemid,
                             const int* __restrict__ ti, const float* __restrict__ tv,
                             int* __restrict__ r, int* __restrict__ c, float* __restrict__ v) {
    int e = blockIdx.x * blockDim.x + threadIdx.x;
    if (e >= E_TOT) return;
    if (e < NNZ) { r[e] = ar[e]; c[e] = ac[e]; v[e] = av[e]; }
    else {
        int j = e - NNZ; int u = j / KK, t = j % KK;
        r[e] = u; c[e] = itemid[ti[u * KK + t]]; v[e] = tv[u * KK + t];
    }
}

__global__ void segsum_kernel(const int* __restrict__ r, const int* __restrict__ c,
                              const float* __restrict__ v, float* __restrict__ sr,
                              float* __restrict__ sc) {
    int e = blockIdx.x * blockDim.x + threadIdx.x;
    if (e >= E_TOT) return;
    atomicAdd(&sr[r[e]], v[e]);
    atomicAdd(&sc[c[e]], v[e]);
}

__global__ void edgenorm_kernel(const int* __restrict__ r, const int* __restrict__ c,
                                const float* __restrict__ v, const float* __restrict__ sr,
                                const float* __restrict__ sc, float* __restrict__ vu,
                                float* __restrict__ vi) {
    int e = blockIdx.x * blockDim.x + threadIdx.x;
    if (e >= E_TOT) return;
    vu[e] = v[e] / sr[r[e]];
    vi[e] = v[e] / sc[c[e]];
}

// out[r[e],:] += v[e] * X[c[e],:]   (out pre-zeroed; edge x col parallel atomics)
__global__ void spmm_kernel(const int* __restrict__ r, const int* __restrict__ c,
                            const float* __restrict__ v, const float* __restrict__ X,
                            float* __restrict__ out, long long total) {
    long long idx = (long long)blockIdx.x * blockDim.x + threadIdx.x;
    if (idx >= total) return;
    int e = (int)(idx >> 9); int hcol = (int)(idx & (Hh - 1));
    atomicAdd(&out[(size_t)r[e] * Hh + hcol], v[e] * X[(size_t)c[e] * Hh + hcol]);
}

__global__ void ew_mul1p_kernel(const float* __restrict__ a, const float* __restrict__ b,
                                float* __restrict__ out, long long n) {
    long long i = (long long)blockIdx.x * blockDim.x + threadIdx.x;
    if (i < n) out[i] = a[i] * (1.0f + b[i]);
}
__global__ void ew_addmul_kernel(float* __restrict__ out, const float* __restrict__ a,
                                 const float* __restrict__ b, long long n) {
    long long i = (long long)blockIdx.x * blockDim.x + threadIdx.x;
    if (i < n) out[i] += a[i] * b[i];
}
__global__ void ew_avg3_kernel(const float* __restrict__ a, const float* __restrict__ b,
                               const float* __restrict__ c, float* __restrict__ out, long long n) {
    long long i = (long long)blockIdx.x * blockDim.x + threadIdx.x;
    if (i < n) out[i] = (a[i] + b[i] + c[i]) * (1.0f / 3.0f);
}

__global__ void inv_init_kernel(int* __restrict__ inv) {
    int i = blockIdx.x * blockDim.x + threadIdx.x;
    if (i < NE) inv[i] = -1;
}
__global__ void inv_set_kernel(const int* __restrict__ itemid, int* __restrict__ inv) {
    int i = blockIdx.x * blockDim.x + threadIdx.x;
    if (i < NI) inv[itemid[i]] = i;
}
__global__ void dense_kernel(const int* __restrict__ ar, const int* __restrict__ ac,
                             const float* __restrict__ av, const int* __restrict__ inv,
                             float* __restrict__ dense) {
    int e = blockIdx.x * blockDim.x + threadIdx.x;
    if (e >= NNZ) return;
    int ic = inv[ac[e]];
    if (ic >= 0) atomicAdd(&dense[(size_t)ar[e] * NI + ic], av[e]);
}
__global__ void loss_kernel(const float* __restrict__ sim, const float* __restrict__ dense,
                            float* __restrict__ loss) {
    __shared__ float red[256];
    const long long total = (long long)NU * NI;
    long long idx = (long long)blockIdx.x * blockDim.x + threadIdx.x;
    const long long stride = (long long)gridDim.x * blockDim.x;
    float s = 0.0f;
    for (long long i = idx; i < total; i += stride) { float d = sim[i] - dense[i]; s += d * d; }
    red[threadIdx.x] = s; __syncthreads();
    for (int st = 128; st > 0; st >>= 1) {
        if (threadIdx.x < st) red[threadIdx.x] += red[threadIdx.x + st];
        __syncthreads();
    }
    if (threadIdx.x == 0) atomicAdd(loss, red[0] * (GAMA / (float)((long long)NU * NI)));
}

// prompt = p2.reshape(B,1,NL,NB,NH,64).transpose(2,3,0,4,1,5)
__global__ void prompt_kernel(const float* __restrict__ p2, float* __restrict__ out) {
    int i = blockIdx.x * blockDim.x + threadIdx.x;
    if (i >= PROMPT_N) return;
    int d  = i & 63;
    int nh = (i >> 6) & 7;
    int b  = (i >> 9) & 63;
    int nb = (i >> 15) & 1;
    int l  = i >> 16;
    out[i] = p2[(size_t)b * P2N + (size_t)(((l * NB + nb) * NH + nh) << 6) + d];
}

// ================= workspace layout (float slots) =================
#define OFF_TOK     ((size_t)0)
#define OFF_GM      (OFF_TOK    + (size_t)Bb*Hh)
#define OFF_FILT    (OFF_GM     + (size_t)Bb*Hh)
#define OFF_HF      (OFF_FILT   + (size_t)Bb*Hh)
#define OFF_UESEL   (OFF_HF     + (size_t)Bb*MD)
#define OFF_ENH     (OFF_UESEL  + (size_t)Bb*Hh)
#define OFF_ALLTOK  (OFF_ENH    + (size_t)Bb*Hh)
#define OFF_TOKEDGE (OFF_ALLTOK + (size_t)NU*Hh)
#define OFF_USERTOK (OFF_TOKEDGE+ (size_t)NU*Hh)
#define OFF_UREP    (OFF_USERTOK+ (size_t)NU*Hh)
#define OFF_IREP    (OFF_UREP   + (size_t)NU*Hh)
#define OFF_SIM     (OFF_IREP   + (size_t)NI*Hh)
#define OFF_TV      (OFF_SIM    + (size_t)NU*NI)
#define OFF_TI      (OFF_TV     + (size_t)NU*KK)
#define OFF_ROWS    (OFF_TI     + (size_t)NU*KK)
#define OFF_COLS    (OFF_ROWS   + (size_t)E_TOT)
#define OFF_VALS    (OFF_COLS   + (size_t)E_TOT)
#define OFF_SUMR    (OFF_VALS   + (size_t)E_TOT)
#define OFF_SUMC    (OFF_SUMR   + (size_t)NU)
#define OFF_VU      (OFF_SUMC   + (size_t)NE)
#define OFF_VI      (OFF_VU     + (size_t)E_TOT)
#define OFF_AI      (OFF_VI     + (size_t)E_TOT)
#define OFF_U1      (OFF_AI     + (size_t)NU*Hh)
#define OFF_U2      (OFF_U1     + (size_t)NU*Hh)
#define OFF_ATT     (OFF_U2     + (size_t)NU*Hh)
#define OFF_I1      (OFF_ATT    + (size_t)NE*Hh)
#define OFF_I2      (OFF_I1     + (size_t)NE*Hh)
#define OFF_UF      (OFF_I2     + (size_t)NE*Hh)
#define OFF_INV     (OFF_UF     + (size_t)NU*Hh)
#define OFF_DENSE   (OFF_INV    + (size_t)NE)
#define OFF_SYSF    (OFF_DENSE  + (size_t)NU*NI)
#define OFF_P1H     (OFF_SYSF   + (size_t)Bb*Hh)
#define OFF_P1      (OFF_P1H    + (size_t)Bb*MD)
#define OFF_P2      (OFF_P1     + (size_t)Bb*Hh)

static inline unsigned cdivu(long long a, long long b) { return (unsigned)((a + b - 1) / b); }

extern "C" void kernel_launch(void* const* d_in, const int* in_sizes, int n_in,
                              void* d_out, int out_size, void* d_ws, size_t ws_size,
                              hipStream_t stream) {
    (void)in_sizes; (void)n_in; (void)out_size; (void)ws_size;
    const int*   systemid  = (const int*)  d_in[0];
    const int*   adj_row   = (const int*)  d_in[1];
    const int*   adj_col   = (const int*)  d_in[2];
    const int*   itemid    = (const int*)  d_in[3];
    const float* token_emb = (const float*)d_in[4];
    const float* adj_val   = (const float*)d_in[5];
    const float* node_emb  = (const float*)d_in[6];
    const float* user_emb  = (const float*)d_in[7];
    const float* tproj_w   = (const float*)d_in[8];
    const float* tproj_b   = (const float*)d_in[9];
    const float* trans_w   = (const float*)d_in[10];
    const float* trans_b   = (const float*)d_in[11];
    const float* trans_t_w = (const float*)d_in[12];
    const float* trans_t_b = (const float*)d_in[13];
    const float* meta_w1   = (const float*)d_in[14];
    const float* meta_b1   = (const float*)d_in[15];
    const float* meta_w2   = (const float*)d_in[16];
    const float* meta_b2   = (const float*)d_in[17];
    const float* sim_w1    = (const float*)d_in[18];
    const float* sim_w2    = (const float*)d_in[19];
    const float* p1_w1     = (const float*)d_in[20];
    const float* p1_b1     = (const float*)d_in[21];
    const float* p1_w2     = (const float*)d_in[22];
    const float* p1_b2     = (const float*)d_in[23];
    const float* p2_w      = (const float*)d_in[24];
    const float* p2_b      = (const float*)d_in[25];
    const float* prelu_a   = (const float*)d_in[26];

    float* ws = (float*)d_ws;
    float* out = (float*)d_out;
    float* out_item = out + PROMPT_N;
    float* out_loss = out + PROMPT_N + (size_t)NE * Hh;

    float* tok   = ws + OFF_TOK;   float* gm    = ws + OFF_GM;
    float* filt  = ws + OFF_FILT;  float* hf    = ws + OFF_HF;
    float* uesel = ws + OFF_UESEL; float* enh   = ws + OFF_ENH;
    float* all_tok = ws + OFF_ALLTOK; float* tok_edge = ws + OFF_TOKEDGE;
    float* user_tok = ws + OFF_USERTOK;
    float* u_rep = ws + OFF_UREP;  float* i_rep = ws + OFF_IREP;
    float* sim   = ws + OFF_SIM;
    float* tv    = ws + OFF_TV;    int* ti = (int*)(ws + OFF_TI);
    int* rows_e  = (int*)(ws + OFF_ROWS); int* cols_e = (int*)(ws + OFF_COLS);
    float* vals_e = ws + OFF_VALS;
    float* sum_r = ws + OFF_SUMR;  float* sum_c = ws + OFF_SUMC;
    float* v_u   = ws + OFF_VU;    float* v_i   = ws + OFF_VI;
    float* Ai    = ws + OFF_AI;    float* u1 = ws + OFF_U1; float* u2 = ws + OFF_U2;
    float* Att   = ws + OFF_ATT;   float* i1 = ws + OFF_I1; float* i2 = ws + OFF_I2;
    float* uf    = ws + OFF_UF;    int* inv = (int*)(ws + OFF_INV);
    float* dense = ws + OFF_DENSE;
    float* sysf  = ws + OFF_SYSF;  float* p1h = ws + OFF_P1H;
    float* p1    = ws + OFF_P1;    float* p2  = ws + OFF_P2;

    const dim3 gblk(128);

    // 1. tok = token_embeds @ token_proj_w + b              [64,512] K=768
    wmma_gemm_kernel<0,0><<<dim3(Hh/64, 1), gblk, 0, stream>>>(
        token_emb, tproj_w, tok, Bb, Hh, TH, TH, Hh, Hh, tproj_b, nullptr, 0, nullptr, 0, nullptr);
    // 2. group mean by systemid
    group_mean_kernel<<<dim3(Bb), dim3(256), 0, stream>>>(systemid, tok, gm);
    // 3. filt = prelu(gm @ trans_w + b) * gm
    wmma_gemm_kernel<0,0><<<dim3(Hh/64, 1), gblk, 0, stream>>>(
        gm, trans_w, filt, Bb, Hh, Hh, Hh, Hh, Hh, trans_b, nullptr, 3, gm, Hh, prelu_a);
    // 4. hf = relu(filt @ meta_w1 + b1)                     [64,256]
    wmma_gemm_kernel<0,0><<<dim3(MD/64, 1), gblk, 0, stream>>>(
        filt, meta_w1, hf, Bb, MD, Hh, Hh, MD, MD, meta_b1, nullptr, 1, nullptr, 0, nullptr);
    // 5. gather ue_sel = user_embedding[systemid]
    gather_rows_kernel<<<dim3(cdivu((long long)Bb*Hh,256)), dim3(256), 0, stream>>>(
        user_emb, systemid, uesel, (long long)Bb*Hh, Hh);
    // 6. fused hypernet -> enhanced (meta_w2 streamed once, mapping never materialized)
    hipMemsetAsync(enh, 0, (size_t)Bb*Hh*sizeof(float), stream);
    hypernet_kernel<<<dim3(N_MAP/64), gblk, 0, stream>>>(hf, meta_w2, meta_b2, uesel, enh);
    // 7. all_tok = user_embedding with scattered enhanced; tok_edge scatter of gm
    hipMemcpyAsync(all_tok, user_emb, (size_t)NU*Hh*sizeof(float), hipMemcpyDeviceToDevice, stream);
    hipMemsetAsync(tok_edge, 0, (size_t)NU*Hh*sizeof(float), stream);
    scatter_rows_kernel<<<dim3(cdivu((long long)Bb*Hh,256)), dim3(256), 0, stream>>>(
        systemid, enh, gm, all_tok, tok_edge);
    // 8. u_rep / i_rep + row-normalize
    wmma_gemm_kernel<0,0><<<dim3(Hh/64, cdivu(NU,64)), gblk, 0, stream>>>(
        all_tok, sim_w1, u_rep, NU, Hh, Hh, Hh, Hh, Hh, nullptr, nullptr, 0, nullptr, 0, nullptr);
    wmma_gemm_kernel<0,1><<<dim3(Hh/64, cdivu(NI,64)), gblk, 0, stream>>>(
        node_emb, sim_w2, i_rep, NI, Hh, Hh, Hh, Hh, Hh, nullptr, itemid, 0, nullptr, 0, nullptr);
    rownorm_kernel<<<dim3(NU), dim3(256), 0, stream>>>(u_rep);
    rownorm_kernel<<<dim3(NI), dim3(256), 0, stream>>>(i_rep);
    // 9. sim = sigmoid(u_n @ i_n^T)                          [2000,4000]
    wmma_gemm_kernel<1,0><<<dim3(cdivu(NI,64), cdivu(NU,64)), gblk, 0, stream>>>(
        u_rep, i_rep, sim, NU, NI, Hh, Hh, Hh, NI, nullptr, nullptr, 2, nullptr, 0, nullptr);
    // 10. top-k per user
    topk_kernel<<<dim3(NU), dim3(256), 0, stream>>>(sim, tv, ti);
    // 11. combined edge list + segment sums + normalization
    edges_kernel<<<dim3(cdivu(E_TOT,256)), dim3(256), 0, stream>>>(
        adj_row, adj_col, adj_val, itemid, ti, tv, rows_e, cols_e, vals_e);
    hipMemsetAsync(sum_r, 0, NU*sizeof(float), stream);
    hipMemsetAsync(sum_c, 0, NE*sizeof(float), stream);
    segsum_kernel<<<dim3(cdivu(E_TOT,256)), dim3(256), 0, stream>>>(rows_e, cols_e, vals_e, sum_r, sum_c);
    edgenorm_kernel<<<dim3(cdivu(E_TOT,256)), dim3(256), 0, stream>>>(
        rows_e, cols_e, vals_e, sum_r, sum_c, v_u, v_i);
    // 12. user_tok = prelu(tok_edge @ trans_t_w + b) * tok_edge
    wmma_gemm_kernel<0,0><<<dim3(Hh/64, cdivu(NU,64)), gblk, 0, stream>>>(
        tok_edge, trans_t_w, user_tok, NU, Hh, Hh, Hh, Hh, Hh, trans_t_b, nullptr, 3, tok_edge, Hh, prelu_a);
    // 13. propagation
    const long long spmm_tot = (long long)E_TOT * Hh;
    const dim3 spmm_grid(cdivu(spmm_tot, 256));
    hipMemsetAsync(Ai, 0, (size_t)NU*Hh*sizeof(float), stream);
    spmm_kernel<<<spmm_grid, dim3(256), 0, stream>>>(rows_e, cols_e, v_u, node_emb, Ai, spmm_tot);
    ew_mul1p_kernel<<<dim3(cdivu((long long)NU*Hh,256)), dim3(256), 0, stream>>>(Ai, user_tok, u1, (long long)NU*Hh);
    hipMemsetAsync(Att, 0, (size_t)NE*Hh*sizeof(float), stream);
    spmm_kernel<<<spmm_grid, dim3(256), 0, stream>>>(cols_e, rows_e, v_i, user_tok, Att, spmm_tot);
    hipMemsetAsync(i1, 0, (size_t)NE*Hh*sizeof(float), stream);
    spmm_kernel<<<spmm_grid, dim3(256), 0, stream>>>(cols_e, rows_e, v_i, all_tok, i1, spmm_tot);
    ew_addmul_kernel<<<dim3(cdivu((long long)NE*Hh,256)), dim3(256), 0, stream>>>(i1, Att, node_emb, (long long)NE*Hh);
    hipMemsetAsync(Ai, 0, (size_t)NU*Hh*sizeof(float), stream);
    spmm_kernel<<<spmm_grid, dim3(256), 0, stream>>>(rows_e, cols_e, v_u, i1, Ai, spmm_tot);
    ew_mul1p_kernel<<<dim3(cdivu((long long)NU*Hh,256)), dim3(256), 0, stream>>>(Ai, user_tok, u2, (long long)NU*Hh);
    hipMemsetAsync(i2, 0, (size_t)NE*Hh*sizeof(float), stream);
    spmm_kernel<<<spmm_grid, dim3(256), 0, stream>>>(cols_e, rows_e, v_i, u1, i2, spmm_tot);
    ew_addmul_kernel<<<dim3(cdivu((long long)NE*Hh,256)), dim3(256), 0, stream>>>(i2, Att, i1, (long long)NE*Hh);
    // 14. finals (item_final straight into d_out)
    ew_avg3_kernel<<<dim3(cdivu((long long)NU*Hh,256)), dim3(256), 0, stream>>>(all_tok, u1, u2, uf, (long long)NU*Hh);
    ew_avg3_kernel<<<dim3(cdivu((long long)NE*Hh,256)), dim3(256), 0, stream>>>(node_emb, i1, i2, out_item, (long long)NE*Hh);
    // 15. loss vs dense reconstruction
    inv_init_kernel<<<dim3(cdivu(NE,256)), dim3(256), 0, stream>>>(inv);
    inv_set_kernel<<<dim3(cdivu(NI,256)), dim3(256), 0, stream>>>(itemid, inv);
    hipMemsetAsync(dense, 0, (size_t)NU*NI*sizeof(float), stream);
    dense_kernel<<<dim3(cdivu(NNZ,256)), dim3(256), 0, stream>>>(adj_row, adj_col, adj_val, inv, dense);
    hipMemsetAsync(out_loss, 0, sizeof(float), stream);
    loss_kernel<<<dim3(8192), dim3(256), 0, stream>>>(sim, dense, out_loss);
    // 16. prompt head
    gather_rows_kernel<<<dim3(cdivu((long long)Bb*Hh,256)), dim3(256), 0, stream>>>(
        uf, systemid, sysf, (long long)Bb*Hh, Hh);
    wmma_gemm_kernel<0,0><<<dim3(MD/64, 1), gblk, 0, stream>>>(
        sysf, p1_w1, p1h, Bb, MD, Hh, Hh, MD, MD, p1_b1, nullptr, 1, nullptr, 0, nullptr);
    wmma_gemm_kernel<0,0><<<dim3(Hh/64, 1), gblk, 0, stream>>>(
        p1h, p1_w2, p1, Bb, Hh, MD, MD, Hh, Hh, p1_b2, nullptr, 4, sysf, Hh, nullptr);
    wmma_gemm_kernel<0,0><<<dim3(P2N/64, 1), gblk, 0, stream>>>(
        p1, p2_w, p2, Bb, P2N, Hh, Hh, P2N, P2N, p2_b, nullptr, 0, nullptr, 0, nullptr);
    prompt_kernel<<<dim3(cdivu(PROMPT_N,256)), dim3(256), 0, stream>>>(p2, out);
}